// PillTeacher_48661979464182
// MI455X (gfx1250) — compile-verified
//
#include <hip/hip_runtime.h>
#include <math.h>

// MPN-COV (iSQRT-COV) + projection head for MI455X (gfx1250).
// All GEMMs on V_WMMA_F32_16X16X4_F32 (full-precision matrix pipe; Newton-
// Schulz is numerically sensitive, so no bf16). NS iterations (19.3 GFLOP,
// L2-resident) use LDS double-buffered 64x128 block tiles with 2x2 register
// blocking per wave; LDS fill goes through GLOBAL_LOAD_ASYNC_TO_LDS_B64
// (ASYNCcnt) -- builtin confirmed present, prototype: (v2i AS1*, v2i AS3*, Ii, Ii).
// Projection streams the 134MB W_proj exactly once (HBM-bound phase).
// Workspace requirement: ~93 MB.

#define ITER_NUM 3
#define BN_EPS 1e-5f
#define L2_EPS 1e-12f

constexpr int B_ = 64;
constexpr int C_ = 256;
constexpr int M_ = 196;       // h*w
constexpr int E_ = 512;
constexpr int CC = C_ * C_;   // 65536

constexpr int KBLK = 32;      // K per LDS stage
constexpr int LDA_S = 36;     // padded LDS row stride (floats), banks disjoint per half-wave
constexpr int LDB_S = 136;    // padded LDS row stride (floats), +16 banks per 2 K-rows
constexpr int A_ELEMS = 64 * LDA_S;    // 2304
constexpr int B_ELEMS = KBLK * LDB_S;  // 4352

typedef __attribute__((ext_vector_type(2))) float v2f;
typedef __attribute__((ext_vector_type(8))) float v8f;
typedef __attribute__((ext_vector_type(2))) int v2i_t;

// ---------------- async global->LDS ---------------------------------------
#if defined(__AMDGCN__) && __has_builtin(__builtin_amdgcn_global_load_async_to_lds_b64)
#define USE_ASYNC 1
#else
#define USE_ASYNC 0
#endif

#if defined(__AMDGCN__) && __has_builtin(__builtin_amdgcn_s_wait_asynccnt)
#define WAIT_ASYNC() __builtin_amdgcn_s_wait_asynccnt(0)
#else
#define WAIT_ASYNC() asm volatile("s_wait_asynccnt 0x0" ::: "memory")
#endif

typedef __attribute__((address_space(1))) v2i_t gas_v2i;
typedef __attribute__((address_space(3))) v2i_t las_v2i;

__device__ __forceinline__ void async_copy_b64(const float* g, float* l) {
#if USE_ASYNC
  __builtin_amdgcn_global_load_async_to_lds_b64(
      (gas_v2i*)(uintptr_t)g,
      (las_v2i*)(uint32_t)(uintptr_t)l,
      0, 0);
#else
  (void)g; (void)l;
#endif
}

__device__ __forceinline__ v8f wmma4(v2f a, v2f b, v8f c) {
  return __builtin_amdgcn_wmma_f32_16x16x4_f32(false, a, false, b, (short)0, c,
                                               false, false);
}

// Direct-from-memory 16x16 tile GEMM (used for covariance, K=196).
// VGPR layout per ISA 7.12.2: A lane l holds M=l, halves split K {k,k+1}/{k+2,k+3};
// C/D: VGPR r, lane (h,l) = element (M=r+8h, N=l).
template <bool BT>
__device__ __forceinline__ v8f wmma_panel(const float* __restrict__ A, int lda,
                                          const float* __restrict__ B, int ldb,
                                          int k0, int k1, v8f acc) {
  const int lane = threadIdx.x & 31;
  const int h = lane >> 4;
  const int l = lane & 15;
  for (int k = k0; k < k1; k += 4) {
    const int kk = k + 2 * h;
    v2f a, b;
    a.x = A[(size_t)l * lda + kk];
    a.y = A[(size_t)l * lda + kk + 1];
    if (BT) {
      b.x = B[(size_t)l * ldb + kk];
      b.y = B[(size_t)l * ldb + kk + 1];
    } else {
      b.x = B[(size_t)kk * ldb + l];
      b.y = B[(size_t)(kk + 1) * ldb + l];
    }
    acc = wmma4(a, b, acc);
  }
  return acc;
}

// ---------------- LDS staging for the NS 256x256x256 GEMMs ----------------
// Block tile: 64 (M) x 128 (N), Kblk = 32. 256 threads.
__device__ __forceinline__ void stage_tiles(const float* __restrict__ Ag, int lda,
                                            const float* __restrict__ Bg, int ldb,
                                            int kb, float* __restrict__ As,
                                            float* __restrict__ Bs) {
  const int tid = threadIdx.x;
#if USE_ASYNC
#pragma unroll
  for (int i = 0; i < 4; ++i) {            // A: 64x32 floats = 1024 b64 chunks
    const int c = tid + i * 256, row = c >> 4, kp = (c & 15) * 2;
    async_copy_b64(Ag + (size_t)row * lda + kb + kp, As + row * LDA_S + kp);
  }
#pragma unroll
  for (int i = 0; i < 8; ++i) {            // B: 32x128 floats = 2048 b64 chunks
    const int c = tid + i * 256, kr = c >> 6, np = (c & 63) * 2;
    async_copy_b64(Bg + (size_t)(kb + kr) * ldb + np, Bs + kr * LDB_S + np);
  }
#else
#pragma unroll
  for (int i = 0; i < 4; ++i) {
    const int c = tid + i * 256, row = c >> 4, kp = (c & 15) * 2;
    *(float2*)&As[row * LDA_S + kp] =
        *(const float2*)&Ag[(size_t)row * lda + kb + kp];
  }
#pragma unroll
  for (int i = 0; i < 8; ++i) {
    const int c = tid + i * 256, kr = c >> 6, np = (c & 63) * 2;
    *(float2*)&Bs[kr * LDB_S + np] =
        *(const float2*)&Bg[(size_t)(kb + kr) * ldb + np];
  }
#endif
}

// Wave computes 32x32 (2x2 tiles of 16x16): 4 WMMAs per 4 LDS loads.
__device__ __forceinline__ void ns_compute_block(const float* __restrict__ As,
                                                 const float* __restrict__ Bs,
                                                 int m0, int n0, v8f& a00,
                                                 v8f& a01, v8f& a10, v8f& a11) {
  const int lane = threadIdx.x & 31;
  const int h = lane >> 4, l = lane & 15;
#pragma unroll
  for (int k = 0; k < KBLK; k += 4) {
    const int kk = k + 2 * h;
    const v2f va0 = *(const v2f*)&As[(m0 + l) * LDA_S + kk];
    const v2f va1 = *(const v2f*)&As[(m0 + 16 + l) * LDA_S + kk];
    v2f vb0, vb1;
    vb0.x = Bs[kk * LDB_S + n0 + l];
    vb0.y = Bs[(kk + 1) * LDB_S + n0 + l];
    vb1.x = Bs[kk * LDB_S + n0 + 16 + l];
    vb1.y = Bs[(kk + 1) * LDB_S + n0 + 16 + l];
    a00 = wmma4(va0, vb0, a00);
    a01 = wmma4(va0, vb1, a01);
    a10 = wmma4(va1, vb0, a10);
    a11 = wmma4(va1, vb1, a11);
  }
}

// Full K=256 pipeline with double-buffered LDS.
__device__ __forceinline__ void ns_gemm(const float* __restrict__ Ag,
                                        const float* __restrict__ Bg,
                                        float (*As)[A_ELEMS], float (*Bs)[B_ELEMS],
                                        int m0, int n0, v8f& a00, v8f& a01,
                                        v8f& a10, v8f& a11) {
  stage_tiles(Ag, C_, Bg, C_, 0, As[0], Bs[0]);
#if USE_ASYNC
  WAIT_ASYNC();
#endif
  __syncthreads();
  int buf = 0;
  for (int kb = 0; kb < C_ / KBLK; ++kb) {
    if (kb + 1 < C_ / KBLK)
      stage_tiles(Ag, C_, Bg, C_, (kb + 1) * KBLK, As[buf ^ 1], Bs[buf ^ 1]);
    ns_compute_block(As[buf], Bs[buf], m0, n0, a00, a01, a10, a11);
#if USE_ASYNC
    WAIT_ASYNC();
#endif
    __syncthreads();
    buf ^= 1;
  }
}

// ---------------- Stage 1: center rows, accumulate trace(S) ---------------
__global__ void center_kernel(const float* __restrict__ feat,
                              float* __restrict__ xc, float* __restrict__ tr) {
  const int b = blockIdx.x;
  const int c = threadIdx.x;
  const float* src = feat + (size_t)(b * C_ + c) * M_;
  float* dst = xc + (size_t)(b * C_ + c) * M_;
  float s = 0.f;
  for (int m = 0; m < M_; ++m) s += src[m];
  const float mean = s * (1.0f / (float)M_);
  float sq = 0.f;
  for (int m = 0; m < M_; ++m) {
    const float v = src[m] - mean;
    dst[m] = v;
    sq += v * v;
  }
  __shared__ float red[256];
  red[c] = sq;
  __syncthreads();
  for (int st = 128; st > 0; st >>= 1) {
    if (c < st) red[c] += red[c + st];
    __syncthreads();
  }
  if (c == 0) tr[b] = red[0];
}

// ---------------- Stage 2: Y0 = (xc*xc^T)/trace(S), Z0 = I ----------------
__global__ void __launch_bounds__(256) cov_init_kernel(
    const float* __restrict__ xc, const float* __restrict__ tr,
    float* __restrict__ Y0, float* __restrict__ Z0) {
  const int b = blockIdx.z;
  const int wave = threadIdx.x >> 5;
  const int m0 = blockIdx.x * 32 + (wave & 1) * 16;
  const int n0 = blockIdx.y * 64 + (wave >> 1) * 16;
  const float* X = xc + (size_t)b * C_ * M_;
  v8f acc = {};
  acc = wmma_panel<true>(X + m0 * M_, M_, X + n0 * M_, M_, 0, M_, acc);
  const int lane = threadIdx.x & 31;
  const int h = lane >> 4, l = lane & 15;
  const float inv = 1.0f / tr[b];
  float* Yp = Y0 + (size_t)b * CC;
  float* Zp = Z0 + (size_t)b * CC;
  for (int r = 0; r < 8; ++r) {
    const int row = m0 + r + 8 * h, col = n0 + l;
    Yp[row * C_ + col] = acc[r] * inv;
    Zp[row * C_ + col] = (row == col) ? 1.f : 0.f;
  }
}

// ---------------- Stage 3a: T = 3I - Z*Y ---------------------------------
__global__ void __launch_bounds__(256) ns_T_kernel(const float* __restrict__ Z,
                                                   const float* __restrict__ Y,
                                                   float* __restrict__ T) {
  __shared__ float As[2][A_ELEMS];
  __shared__ float Bs[2][B_ELEMS];
  const int b = blockIdx.z;
  const float* Ag = Z + (size_t)b * CC + (size_t)blockIdx.x * 64 * C_;
  const float* Bg = Y + (size_t)b * CC + (size_t)blockIdx.y * 128;
  const int wave = threadIdx.x >> 5;
  const int m0 = (wave & 1) * 32, n0 = (wave >> 1) * 32;
  v8f a00 = {}, a01 = {}, a10 = {}, a11 = {};
  ns_gemm(Ag, Bg, As, Bs, m0, n0, a00, a01, a10, a11);
  const int lane = threadIdx.x & 31;
  const int h = lane >> 4, l = lane & 15;
  float* Tp = T + (size_t)b * CC;
  const int rbase = blockIdx.x * 64 + m0;
  const int cbase = blockIdx.y * 128 + n0;
  const v8f* accs[4] = {&a00, &a01, &a10, &a11};
  for (int t = 0; t < 4; ++t) {
    const int r0 = rbase + (t >> 1) * 16;
    const int c0 = cbase + (t & 1) * 16 + l;
    for (int r = 0; r < 8; ++r) {
      const int row = r0 + r + 8 * h;
      Tp[row * C_ + c0] = ((row == c0) ? 3.f : 0.f) - (*accs[t])[r];
    }
  }
}

// ---------------- Stage 3b: Yn = 0.5*Y*T  /  Zn = 0.5*T*Z -----------------
__global__ void __launch_bounds__(256) ns_YZ_kernel(
    const float* __restrict__ Y, const float* __restrict__ Z,
    const float* __restrict__ T, float* __restrict__ Yn,
    float* __restrict__ Zn) {
  __shared__ float As[2][A_ELEMS];
  __shared__ float Bs[2][B_ELEMS];
  const int bz = blockIdx.z;
  const int b = bz & (B_ - 1);
  const int which = bz >> 6;  // block-uniform: EXEC stays full
  const float* Ap;
  const float* Bp;
  float* Op;
  if (which == 0) { Ap = Y; Bp = T; Op = Yn; }   // Yn = 0.5*Y*T
  else            { Ap = T; Bp = Z; Op = Zn; }   // Zn = 0.5*T*Z
  const float* Ag = Ap + (size_t)b * CC + (size_t)blockIdx.x * 64 * C_;
  const float* Bg = Bp + (size_t)b * CC + (size_t)blockIdx.y * 128;
  const int wave = threadIdx.x >> 5;
  const int m0 = (wave & 1) * 32, n0 = (wave >> 1) * 32;
  v8f a00 = {}, a01 = {}, a10 = {}, a11 = {};
  ns_gemm(Ag, Bg, As, Bs, m0, n0, a00, a01, a10, a11);
  const int lane = threadIdx.x & 31;
  const int h = lane >> 4, l = lane & 15;
  float* Out = Op + (size_t)b * CC;
  const int rbase = blockIdx.x * 64 + m0;
  const int cbase = blockIdx.y * 128 + n0;
  const v8f* accs[4] = {&a00, &a01, &a10, &a11};
  for (int t = 0; t < 4; ++t) {
    const int r0 = rbase + (t >> 1) * 16;
    const int c0 = cbase + (t & 1) * 16 + l;
    for (int r = 0; r < 8; ++r) {
      const int row = r0 + r + 8 * h;
      Out[row * C_ + c0] = 0.5f * (*accs[t])[r];
    }
  }
}

// ---------------- Stage 4: per-batch flat L2 scale alpha ------------------
__global__ void rownorm_alpha_kernel(const float* __restrict__ Yf,
                                     const float* __restrict__ tr,
                                     float* __restrict__ alpha) {
  const int b = blockIdx.x;
  const int t = threadIdx.x;
  const float* p = Yf + (size_t)b * CC;
  float s = 0.f;
  for (int i = t; i < CC; i += 256) {
    const float v = p[i];
    s += v * v;
  }
  __shared__ float red[256];
  red[t] = s;
  __syncthreads();
  for (int st = 128; st > 0; st >>= 1) {
    if (t < st) red[t] += red[t + st];
    __syncthreads();
  }
  if (t == 0) {
    const float trY = tr[b] * (1.0f / (float)M_);
    const float sc = sqrtf(trY);
    const float nrm = fmaxf(sc * sqrtf(red[0]), L2_EPS);
    alpha[b] = sc / nrm;
  }
}

__global__ void zero_emb_kernel(float* __restrict__ emb) {
  const int i = blockIdx.x * 256 + threadIdx.x;
  if (i < B_ * E_) emb[i] = 0.f;
}

// ---------------- Stage 5: projection GEMM [64x65536]x[65536x512] ---------
// One block column covers ALL 64 batch rows -> W_proj streamed exactly once.
// Per wave: 4 M-tiles x 1 N-tile = 4 WMMAs per 5 loads. K split over grid.y.
__global__ void __launch_bounds__(256) proj_kernel(
    const float* __restrict__ Yf, const float* __restrict__ W,
    const float* __restrict__ alpha, float* __restrict__ emb) {
  const int KS = CC / 32;  // 2048
  const int wave = threadIdx.x >> 5;
  const int n0 = blockIdx.x * 128 + wave * 16;  // grid.x = 4 -> 512 cols
  const int k0 = blockIdx.y * KS;               // grid.y = 32
  const int lane = threadIdx.x & 31;
  const int h = lane >> 4, l = lane & 15;
  v8f acc0 = {}, acc1 = {}, acc2 = {}, acc3 = {};
  const float* Wr = W + (size_t)n0 * CC;
  for (int k = k0; k < k0 + KS; k += 4) {
    const int kk = k + 2 * h;
    v2f bv;
    bv.x = Wr[(size_t)l * CC + kk];
    bv.y = Wr[(size_t)l * CC + kk + 1];
    const v2f a0 = *(const v2f*)&Yf[(size_t)(l) * CC + kk];
    const v2f a1 = *(const v2f*)&Yf[(size_t)(16 + l) * CC + kk];
    const v2f a2 = *(const v2f*)&Yf[(size_t)(32 + l) * CC + kk];
    const v2f a3 = *(const v2f*)&Yf[(size_t)(48 + l) * CC + kk];
    acc0 = wmma4(a0, bv, acc0);
    acc1 = wmma4(a1, bv, acc1);
    acc2 = wmma4(a2, bv, acc2);
    acc3 = wmma4(a3, bv, acc3);
  }
  const v8f* accs[4] = {&acc0, &acc1, &acc2, &acc3};
  for (int t = 0; t < 4; ++t) {
    for (int r = 0; r < 8; ++r) {
      const int row = t * 16 + r + 8 * h;
      atomicAdd(&emb[row * E_ + n0 + l], (*accs[t])[r] * alpha[row]);
    }
  }
}

// ---------------- Stage 6: bias + BN(inference) + final L2 norm -----------
__global__ void finish_kernel(const float* __restrict__ emb,
                              const float* __restrict__ b_proj,
                              const float* __restrict__ gamma,
                              const float* __restrict__ beta,
                              const float* __restrict__ mean,
                              const float* __restrict__ var,
                              float* __restrict__ out) {
  const int b = blockIdx.x;
  const int e = threadIdx.x;  // 512 threads
  float v = emb[b * E_ + e] + b_proj[e];
  v = (v - mean[e]) * rsqrtf(var[e] + BN_EPS) * gamma[e] + beta[e];
  __shared__ float red[512];
  red[e] = v * v;
  __syncthreads();
  for (int st = 256; st > 0; st >>= 1) {
    if (e < st) red[e] += red[e + st];
    __syncthreads();
  }
  const float nrm = fmaxf(sqrtf(red[0]), L2_EPS);
  out[b * E_ + e] = v / nrm;
}

extern "C" void kernel_launch(void* const* d_in, const int* in_sizes, int n_in,
                              void* d_out, int out_size, void* d_ws,
                              size_t ws_size, hipStream_t stream) {
  (void)in_sizes; (void)n_in; (void)out_size; (void)ws_size;
  const float* feat = (const float*)d_in[0];
  const float* W    = (const float*)d_in[1];
  const float* bpr  = (const float*)d_in[2];
  const float* gam  = (const float*)d_in[3];
  const float* bet  = (const float*)d_in[4];
  const float* mea  = (const float*)d_in[5];
  const float* var  = (const float*)d_in[6];
  float* out = (float*)d_out;

  char* ws = (char*)d_ws;
  size_t off = 0;
  auto alloc = [&](size_t nfloats) -> float* {
    float* p = (float*)(ws + off);
    off += ((nfloats * sizeof(float) + 255) / 256) * 256;
    return p;
  };
  float* XC    = alloc((size_t)B_ * C_ * M_);
  float* Ya    = alloc((size_t)B_ * CC);
  float* Za    = alloc((size_t)B_ * CC);
  float* Yb    = alloc((size_t)B_ * CC);
  float* Zb    = alloc((size_t)B_ * CC);
  float* Tt    = alloc((size_t)B_ * CC);
  float* tr    = alloc(B_);
  float* alpha = alloc(B_);
  float* emb   = alloc((size_t)B_ * E_);

  center_kernel<<<B_, 256, 0, stream>>>(feat, XC, tr);
  cov_init_kernel<<<dim3(8, 4, B_), 256, 0, stream>>>(XC, tr, Ya, Za);

  float* Yc = Ya; float* Zc = Za; float* Yn = Yb; float* Zn = Zb;
  for (int i = 0; i < ITER_NUM; ++i) {
    ns_T_kernel<<<dim3(4, 2, B_), 256, 0, stream>>>(Zc, Yc, Tt);
    ns_YZ_kernel<<<dim3(4, 2, 2 * B_), 256, 0, stream>>>(Yc, Zc, Tt, Yn, Zn);
    float* t1 = Yc; Yc = Yn; Yn = t1;
    float* t2 = Zc; Zc = Zn; Zn = t2;
  }

  rownorm_alpha_kernel<<<B_, 256, 0, stream>>>(Yc, tr, alpha);
  zero_emb_kernel<<<(B_ * E_ + 255) / 256, 256, 0, stream>>>(emb);
  proj_kernel<<<dim3(E_ / 128, 32), 256, 0, stream>>>(Yc, W, alpha, emb);
  finish_kernel<<<B_, E_, 0, stream>>>(emb, bpr, gam, bet, mea, var, out);
}